// DeepFactorizationMachineModel_7610682048652
// MI455X (gfx1250) — compile-verified
//
#include <hip/hip_runtime.h>

typedef __attribute__((ext_vector_type(2))) float v2f;
typedef __attribute__((ext_vector_type(8))) float v8f;

#define HSTR 260   // 256 + 4-float pad: A-column reads hit 16 distinct LDS banks
#define H1S   68   // 64 + 4 pad
#define H2S   36   // 32 + 4 pad

__global__ __launch_bounds__(128)
void dfm_fused_kernel(const int*   __restrict__ x_field,
                      const int*   __restrict__ x,
                      const float* __restrict__ x_val,
                      const float* __restrict__ emb,
                      const float* __restrict__ lin_tab,
                      const float* __restrict__ lin_bias,
                      const float* __restrict__ W1, const float* __restrict__ b1,
                      const float* __restrict__ g1, const float* __restrict__ be1,
                      const float* __restrict__ W2, const float* __restrict__ b2,
                      const float* __restrict__ g2, const float* __restrict__ be2,
                      const float* __restrict__ W3, const float* __restrict__ b3,
                      float*       __restrict__ out)
{
    __shared__ float sh_h [16 * HSTR];  // 16x256 activation tile (embed_x rows)
    __shared__ float sh_h1[16 * H1S];   // layer-1 output 16x64
    __shared__ float sh_h2[16 * H2S];   // layer-2 output 16x32
    __shared__ float sh_red[16 * 8];    // FM partial reduction
    __shared__ float sh_lr[256];        // per-(row,field) linear partials
    __shared__ float sh_lf[16];         // per-row lin + fm

    const int t    = threadIdx.x;
    const int wave = t >> 5;
    const int lane = t & 31;
    const int half = lane >> 4;     // 0: lanes 0-15, 1: lanes 16-31
    const int mlo  = lane & 15;     // A-fragment row / B-fragment col index
    const int r0   = blockIdx.x * 16;
    const float RSQ = rsqrtf(1.0f + 1e-5f);   // BN eval-mode scale, folded

    // ---- phase 1: gather 16 rows x 16 fields x 16 dims, value-weighted ----
    for (int p = t; p < 256; p += 128) {
        const int r = p >> 4, f = p & 15;
        const int gi = (r0 + r) * 16 + f;
        const long long row = (long long)x_field[gi] * 100000ll + (long long)x[gi];
        const float val = x_val[gi];
        const float4* e4 = reinterpret_cast<const float4*>(emb + (size_t)row * 16u);
        float4 a0 = e4[0], a1 = e4[1], a2 = e4[2], a3 = e4[3];
        float4* dst = reinterpret_cast<float4*>(&sh_h[r * HSTR + f * 16]);
        dst[0] = make_float4(a0.x * val, a0.y * val, a0.z * val, a0.w * val);
        dst[1] = make_float4(a1.x * val, a1.y * val, a1.z * val, a1.w * val);
        dst[2] = make_float4(a2.x * val, a2.y * val, a2.z * val, a2.w * val);
        dst[3] = make_float4(a3.x * val, a3.y * val, a3.z * val, a3.w * val);
        sh_lr[p] = lin_tab[row] * val;
    }
    __syncthreads();

    // ---- phase 2: FM term 0.5*sum_e((sum_f v)^2 - sum_f v^2), + linear ----
    {
        const int r = t >> 3;
        const int e = (t & 7) * 2;
        float s0 = 0.f, q0 = 0.f, s1 = 0.f, q1 = 0.f;
        #pragma unroll
        for (int f = 0; f < 16; ++f) {
            const float v0 = sh_h[r * HSTR + f * 16 + e];
            const float v1 = sh_h[r * HSTR + f * 16 + e + 1];
            s0 += v0; q0 = fmaf(v0, v0, q0);
            s1 += v1; q1 = fmaf(v1, v1, q1);
        }
        sh_red[r * 8 + (t & 7)] = (s0 * s0 - q0) + (s1 * s1 - q1);
    }
    __syncthreads();
    if (t < 16) {
        float fm = 0.f;
        #pragma unroll
        for (int i = 0; i < 8; ++i) fm += sh_red[t * 8 + i];
        float lin = 0.f;
        #pragma unroll
        for (int f = 0; f < 16; ++f) lin += sh_lr[t * 16 + f];
        sh_lf[t] = lin + lin_bias[0] + 0.5f * fm;
    }
    __syncthreads();

    // ---- phase 3: layer 1  h(16x256) @ W1(256x64), fp32 WMMA 16x16x4 ----
    // Each wave owns a 16x16 output tile (cols 16*wave .. +15).
    {
        const int c = wave * 16 + mlo;
        v8f acc = {};
        #pragma unroll 4
        for (int kk = 0; kk < 256; kk += 4) {
            const int k0 = kk + 2 * half;                       // lanes0-15: K{kk,kk+1}, lanes16-31: K{kk+2,kk+3}
            v2f a = *(const v2f*)&sh_h[mlo * HSTR + k0];        // ds_load_b64
            v2f b; b.x = W1[(size_t)k0 * 64 + c];
                   b.y = W1[(size_t)(k0 + 1) * 64 + c];
            acc = __builtin_amdgcn_wmma_f32_16x16x4_f32(false, a, false, b,
                                                        (short)0, acc, false, false);
        }
        // BN(eval) + ReLU epilogue: (acc + b1)*g1/sqrt(1+eps) + be1
        const float sc = g1[c] * RSQ;
        const float tb = fmaf(b1[c], sc, be1[c]);
        #pragma unroll
        for (int j = 0; j < 8; ++j) {
            const int m = j + 8 * half;                         // C/D layout: VGPR j -> M=j / M=j+8
            const float v = fmaf(acc[j], sc, tb);
            sh_h1[m * H1S + c] = v > 0.f ? v : 0.f;
        }
    }
    __syncthreads();

    // ---- phase 4: layer 2  h1(16x64) @ W2(64x32), 2 waves ----
    if (wave < 2) {
        const int c = wave * 16 + mlo;
        v8f acc = {};
        #pragma unroll 4
        for (int kk = 0; kk < 64; kk += 4) {
            const int k0 = kk + 2 * half;
            v2f a = *(const v2f*)&sh_h1[mlo * H1S + k0];
            v2f b; b.x = W2[k0 * 32 + c];
                   b.y = W2[(k0 + 1) * 32 + c];
            acc = __builtin_amdgcn_wmma_f32_16x16x4_f32(false, a, false, b,
                                                        (short)0, acc, false, false);
        }
        const float sc = g2[c] * RSQ;
        const float tb = fmaf(b2[c], sc, be2[c]);
        #pragma unroll
        for (int j = 0; j < 8; ++j) {
            const int m = j + 8 * half;
            const float v = fmaf(acc[j], sc, tb);
            sh_h2[m * H2S + c] = v > 0.f ? v : 0.f;
        }
    }
    __syncthreads();

    // ---- phase 5: layer 3 (32->1) + lin + fm -> output ----
    if (t < 16) {
        float acc = sh_lf[t] + b3[0];
        #pragma unroll
        for (int cc = 0; cc < 32; ++cc)
            acc = fmaf(sh_h2[t * H2S + cc], W3[cc], acc);
        out[r0 + t] = acc;
    }
}

extern "C" void kernel_launch(void* const* d_in, const int* in_sizes, int n_in,
                              void* d_out, int out_size, void* d_ws, size_t ws_size,
                              hipStream_t stream) {
    const int*   x_field  = (const int*)  d_in[0];
    const int*   x        = (const int*)  d_in[1];
    const float* x_val    = (const float*)d_in[2];
    const float* emb      = (const float*)d_in[3];
    const float* lin_tab  = (const float*)d_in[4];
    const float* lin_bias = (const float*)d_in[5];
    const float* W1 = (const float*)d_in[6];
    const float* b1 = (const float*)d_in[7];
    const float* g1 = (const float*)d_in[8];
    const float* be1= (const float*)d_in[9];
    const float* W2 = (const float*)d_in[10];
    const float* b2 = (const float*)d_in[11];
    const float* g2 = (const float*)d_in[12];
    const float* be2= (const float*)d_in[13];
    const float* W3 = (const float*)d_in[14];
    const float* b3 = (const float*)d_in[15];
    float* out = (float*)d_out;

    const int nblocks = out_size / 16;   // 16384 / 16 = 1024 workgroups of 16 rows
    dfm_fused_kernel<<<dim3(nblocks), dim3(128), 0, stream>>>(
        x_field, x, x_val, emb, lin_tab, lin_bias,
        W1, b1, g1, be1, W2, b2, g2, be2, W3, b3, out);
}